// MultiHeadSelfAttention_49632642072648
// MI455X (gfx1250) — compile-verified
//
#include <hip/hip_runtime.h>

typedef __attribute__((ext_vector_type(2))) float v2f;
typedef __attribute__((ext_vector_type(8))) float v8f;

// D = A(16x4) * B(4x16) + C, full fp32
#define WMMA4(a, b, c) \
  __builtin_amdgcn_wmma_f32_16x16x4_f32(false, (a), false, (b), (short)0, (c), false, false)

namespace {
constexpr int S = 2048;    // sequence length
constexpr int D = 1024;    // d_model
constexpr int H = 16;      // heads
constexpr int DK = 64;     // head dim
constexpr int M = 2 * S;   // B*S = 4096 rows
constexpr int QKV_N = 3 * D;
}

// ---------------------------------------------------------------------------
// Kernel 1: QKV projection  qkv[m,e] = sum_d x[m,d] * Wqkv[e,d]  (NT GEMM)
// scatter epilogue into Q/K/V workspace laid out [B*H, S, 64]
// Block = 256 threads (8 waves); wave tile = 32x64 (2x4 subtiles).
// ---------------------------------------------------------------------------
__global__ __launch_bounds__(256) void qkv_gemm(const float* __restrict__ X,
                                                const float* __restrict__ W,
                                                float* __restrict__ qp,
                                                float* __restrict__ kp,
                                                float* __restrict__ vp) {
  const int lane = threadIdx.x & 31;
  const int wave = threadIdx.x >> 5;
  const int r = lane & 15, hk = lane >> 4;
  const int rowBase = blockIdx.y * 128 + (wave & 3) * 32;   // M tile of this wave
  const int colBase = blockIdx.x * 128 + (wave >> 2) * 64;  // N tile of this wave

  const float* a0 = X + (rowBase + r) * D + 2 * hk;
  const float* a1 = a0 + 16 * D;
  const float* b0 = W + (colBase + r) * D + 2 * hk;
  const float* b1 = b0 + 16 * D;
  const float* b2 = b0 + 32 * D;
  const float* b3 = b0 + 48 * D;

  v8f acc[2][4] = {};
#pragma unroll 4
  for (int k = 0; k < D; k += 4) {
    v2f A0 = *(const v2f*)(a0 + k);
    v2f A1 = *(const v2f*)(a1 + k);
    v2f B0 = *(const v2f*)(b0 + k);
    v2f B1 = *(const v2f*)(b1 + k);
    v2f B2 = *(const v2f*)(b2 + k);
    v2f B3 = *(const v2f*)(b3 + k);
    acc[0][0] = WMMA4(A0, B0, acc[0][0]);
    acc[0][1] = WMMA4(A0, B1, acc[0][1]);
    acc[0][2] = WMMA4(A0, B2, acc[0][2]);
    acc[0][3] = WMMA4(A0, B3, acc[0][3]);
    acc[1][0] = WMMA4(A1, B0, acc[1][0]);
    acc[1][1] = WMMA4(A1, B1, acc[1][1]);
    acc[1][2] = WMMA4(A1, B2, acc[1][2]);
    acc[1][3] = WMMA4(A1, B3, acc[1][3]);
  }

  // epilogue: e -> (which, head, dk); row -> (b, s)
#pragma unroll
  for (int i = 0; i < 2; ++i) {
#pragma unroll
    for (int j = 0; j < 4; ++j) {
      const int col = colBase + j * 16 + r;
      const int which = col >> 10;
      const int h = (col >> 6) & 15;
      const int dkk = col & 63;
      float* base = (which == 0) ? qp : (which == 1) ? kp : vp;
#pragma unroll
      for (int v = 0; v < 8; ++v) {
        const int row = rowBase + i * 16 + 8 * hk + v;
        const int b = row >> 11;
        const int s = row & (S - 1);
        base[(((b * H + h) * S + s) << 6) + dkk] = acc[i][j][v];
      }
    }
  }
}

// ---------------------------------------------------------------------------
// Kernel 2: interleaved RoPE in place on Q and K.  One thread per (bh, s, t)
// rotation pair; sincos shared between Q and K.
// ---------------------------------------------------------------------------
__global__ __launch_bounds__(256) void rope_kernel(float* __restrict__ qp,
                                                   float* __restrict__ kp,
                                                   const int* __restrict__ tpos) {
  const int idx = blockIdx.x * blockDim.x + threadIdx.x;  // [0, 32*2048*32)
  const int t = idx & 31;
  const int s = (idx >> 5) & (S - 1);
  const int bh = idx >> 16;
  if (bh >= 2 * H) return;
  const int b = bh >> 4;

  const float pos = (float)tpos[b * S + s];
  // freq = theta^(-2t/64) = exp(-t * (1/32) * ln(10000))
  const float freq = __expf(-(float)t * 0.03125f * 9.210340372f);
  float sn, cs;
  __sincosf(pos * freq, &sn, &cs);

  const size_t off = ((size_t)(bh * S + s) << 6) + 2 * t;
  const float qe = qp[off], qo = qp[off + 1];
  qp[off]     = cs * qe - sn * qo;
  qp[off + 1] = sn * qe + cs * qo;
  const float ke = kp[off], ko = kp[off + 1];
  kp[off]     = cs * ke - sn * ko;
  kp[off + 1] = sn * ke + cs * ko;
}

// 16-lane reductions (rows live within one half-wave in the C layout;
// xor masks < 16 never cross the half boundary in wave32)
__device__ __forceinline__ float rowMax16(float x) {
  x = fmaxf(x, __shfl_xor(x, 8));
  x = fmaxf(x, __shfl_xor(x, 4));
  x = fmaxf(x, __shfl_xor(x, 2));
  x = fmaxf(x, __shfl_xor(x, 1));
  return x;
}
__device__ __forceinline__ float rowSum16(float x) {
  x += __shfl_xor(x, 8);
  x += __shfl_xor(x, 4);
  x += __shfl_xor(x, 2);
  x += __shfl_xor(x, 1);
  return x;
}

// ---------------------------------------------------------------------------
// Kernel 3: causal flash attention. One wave per (head, 32-row q tile).
// Q held in registers (pre-scaled by 1/sqrt(dk)); online softmax; P goes
// through a 4KB LDS tile to convert C-layout -> A-layout for P@V.
// ---------------------------------------------------------------------------
__global__ __launch_bounds__(32) void attn_kernel(const float* __restrict__ Q,
                                                  const float* __restrict__ K,
                                                  const float* __restrict__ V,
                                                  float* __restrict__ O) {
  __shared__ float pTile[32 * 32];

  const int bh = blockIdx.y;          // 0..31
  const int q0 = blockIdx.x * 32;     // q tile start
  const int lane = threadIdx.x;
  const int r = lane & 15, hk = lane >> 4;

  const float* Qb = Q + (size_t)bh * S * DK;
  const float* Kb = K + (size_t)bh * S * DK;
  const float* Vb = V + (size_t)bh * S * DK;

  // Q fragments: qa[i][ks] covers rows q0+16i..+15, k = 4*ks + 2*hk + {0,1}
  v2f qa[2][16];
#pragma unroll
  for (int i = 0; i < 2; ++i) {
    const float* qrow = Qb + (q0 + i * 16 + r) * DK + 2 * hk;
#pragma unroll
    for (int ks = 0; ks < 16; ++ks) {
      v2f q = *(const v2f*)(qrow + ks * 4);
      qa[i][ks] = q * 0.125f;  // 1/sqrt(64)
    }
  }

  v8f o[2][4] = {};
  float mrow[2][8], lrow[2][8];
#pragma unroll
  for (int i = 0; i < 2; ++i)
#pragma unroll
    for (int v = 0; v < 8; ++v) { mrow[i][v] = -3.0e38f; lrow[i][v] = 0.0f; }

  const int nkv = (q0 >> 5) + 1;  // causal: kv tiles 0..q-tile inclusive
  for (int t = 0; t < nkv; ++t) {
    const int kv0 = t * 32;

    // ---- scores = Q @ K^T (32x32 tile) ----
    v8f sc[2][2] = {};
#pragma unroll
    for (int ks = 0; ks < 16; ++ks) {
      v2f kb0 = *(const v2f*)(Kb + (kv0 + r) * DK + ks * 4 + 2 * hk);
      v2f kb1 = *(const v2f*)(Kb + (kv0 + 16 + r) * DK + ks * 4 + 2 * hk);
      sc[0][0] = WMMA4(qa[0][ks], kb0, sc[0][0]);
      sc[0][1] = WMMA4(qa[0][ks], kb1, sc[0][1]);
      sc[1][0] = WMMA4(qa[1][ks], kb0, sc[1][0]);
      sc[1][1] = WMMA4(qa[1][ks], kb1, sc[1][1]);
    }

    // ---- causal mask (only diagonal tile is partial) ----
    if (kv0 == q0) {
#pragma unroll
      for (int i = 0; i < 2; ++i)
#pragma unroll
        for (int j = 0; j < 2; ++j)
#pragma unroll
          for (int v = 0; v < 8; ++v) {
            const int qr = 16 * i + 8 * hk + v;
            const int kc = 16 * j + r;
            if (kc > qr) sc[i][j][v] = -1.0e30f;
          }
    }

    // ---- online softmax ----
#pragma unroll
    for (int i = 0; i < 2; ++i) {
#pragma unroll
      for (int v = 0; v < 8; ++v) {
        float mx = rowMax16(fmaxf(sc[0 + i][0][v], sc[i][1][v]));
        const float mnew = fmaxf(mrow[i][v], mx);
        const float alpha = __expf(mrow[i][v] - mnew);
        mrow[i][v] = mnew;
        const float p0 = __expf(sc[i][0][v] - mnew);
        const float p1 = __expf(sc[i][1][v] - mnew);
        sc[i][0][v] = p0;
        sc[i][1][v] = p1;
        lrow[i][v] = lrow[i][v] * alpha + rowSum16(p0 + p1);
#pragma unroll
        for (int n = 0; n < 4; ++n) o[i][n][v] *= alpha;
      }
    }

    // ---- P: C-layout -> A-layout via LDS ----
#pragma unroll
    for (int i = 0; i < 2; ++i)
#pragma unroll
      for (int j = 0; j < 2; ++j)
#pragma unroll
        for (int v = 0; v < 8; ++v)
          pTile[(16 * i + 8 * hk + v) * 32 + 16 * j + r] = sc[i][j][v];
    __syncthreads();

    // ---- O += P @ V ----
#pragma unroll
    for (int ks = 0; ks < 8; ++ks) {
      v2f pa0 = *(const v2f*)(&pTile[r * 32 + ks * 4 + 2 * hk]);
      v2f pa1 = *(const v2f*)(&pTile[(16 + r) * 32 + ks * 4 + 2 * hk]);
      const int kk = kv0 + ks * 4 + 2 * hk;
#pragma unroll
      for (int n = 0; n < 4; ++n) {
        v2f vb;
        vb.x = Vb[(kk + 0) * DK + n * 16 + r];
        vb.y = Vb[(kk + 1) * DK + n * 16 + r];
        o[0][n] = WMMA4(pa0, vb, o[0][n]);
        o[1][n] = WMMA4(pa1, vb, o[1][n]);
      }
    }
    __syncthreads();
  }

  // ---- normalize and write to attention buffer [B, S, D] ----
  const int b = bh >> 4, h = bh & 15;
#pragma unroll
  for (int i = 0; i < 2; ++i)
#pragma unroll
    for (int n = 0; n < 4; ++n)
#pragma unroll
      for (int v = 0; v < 8; ++v) {
        const int srow = q0 + 16 * i + 8 * hk + v;
        O[((size_t)(b * S + srow)) * D + h * DK + n * 16 + r] = o[i][n][v] / lrow[i][v];
      }
}

// ---------------------------------------------------------------------------
// Kernel 4: output projection  out[m,e] = sum_d ao[m,d] * Wo[e,d]  (NT GEMM)
// ---------------------------------------------------------------------------
__global__ __launch_bounds__(256) void out_gemm(const float* __restrict__ X,
                                                const float* __restrict__ W,
                                                float* __restrict__ out) {
  const int lane = threadIdx.x & 31;
  const int wave = threadIdx.x >> 5;
  const int r = lane & 15, hk = lane >> 4;
  const int rowBase = blockIdx.y * 128 + (wave & 3) * 32;
  const int colBase = blockIdx.x * 128 + (wave >> 2) * 64;

  const float* a0 = X + (rowBase + r) * D + 2 * hk;
  const float* a1 = a0 + 16 * D;
  const float* b0 = W + (colBase + r) * D + 2 * hk;
  const float* b1 = b0 + 16 * D;
  const float* b2 = b0 + 32 * D;
  const float* b3 = b0 + 48 * D;

  v8f acc[2][4] = {};
#pragma unroll 4
  for (int k = 0; k < D; k += 4) {
    v2f A0 = *(const v2f*)(a0 + k);
    v2f A1 = *(const v2f*)(a1 + k);
    v2f B0 = *(const v2f*)(b0 + k);
    v2f B1 = *(const v2f*)(b1 + k);
    v2f B2 = *(const v2f*)(b2 + k);
    v2f B3 = *(const v2f*)(b3 + k);
    acc[0][0] = WMMA4(A0, B0, acc[0][0]);
    acc[0][1] = WMMA4(A0, B1, acc[0][1]);
    acc[0][2] = WMMA4(A0, B2, acc[0][2]);
    acc[0][3] = WMMA4(A0, B3, acc[0][3]);
    acc[1][0] = WMMA4(A1, B0, acc[1][0]);
    acc[1][1] = WMMA4(A1, B1, acc[1][1]);
    acc[1][2] = WMMA4(A1, B2, acc[1][2]);
    acc[1][3] = WMMA4(A1, B3, acc[1][3]);
  }

#pragma unroll
  for (int i = 0; i < 2; ++i)
#pragma unroll
    for (int j = 0; j < 4; ++j) {
      const int col = colBase + j * 16 + r;
#pragma unroll
      for (int v = 0; v < 8; ++v) {
        const int row = rowBase + i * 16 + 8 * hk + v;
        out[(size_t)row * D + col] = acc[i][j][v];
      }
    }
}

// ---------------------------------------------------------------------------
extern "C" void kernel_launch(void* const* d_in, const int* in_sizes, int n_in,
                              void* d_out, int out_size, void* d_ws, size_t ws_size,
                              hipStream_t stream) {
  const float* x    = (const float*)d_in[0];   // [B,S,D]
  const int*   tpos = (const int*)d_in[1];     // [B,S]
  const float* wqkv = (const float*)d_in[2];   // [3D, D]
  const float* wo   = (const float*)d_in[3];   // [D, D]
  float* out = (float*)d_out;                  // [B,S,D]

  const size_t planeElems = (size_t)2 * H * S * DK;  // 4M floats = 16MB
  float* qp = (float*)d_ws;
  float* kp = qp + planeElems;
  float* vp = kp + planeElems;
  float* ao = vp + planeElems;                 // attention output [B,S,D]

  // 1) QKV projection (NT GEMM, WMMA f32): grid covers 4096 x 3072
  qkv_gemm<<<dim3(QKV_N / 128, M / 128), 256, 0, stream>>>(x, wqkv, qp, kp, vp);

  // 2) RoPE on Q and K: 32 bh * 2048 s * 32 pairs = 2M threads
  rope_kernel<<<dim3((2 * H * S * 32) / 256), 256, 0, stream>>>(qp, kp, tpos);

  // 3) Flash attention: one wave per (bh, 32-row q tile)
  attn_kernel<<<dim3(S / 32, 2 * H), 32, 0, stream>>>(qp, kp, vp, ao);

  // 4) Output projection into d_out
  out_gemm<<<dim3(D / 128, M / 128), 256, 0, stream>>>(ao, wo, out);
}